// KPlanesEncoder_33474975105277
// MI455X (gfx1250) — compile-verified
//
#include <hip/hip_runtime.h>

// K-Planes encoder for MI455X (gfx1250, wave32).
// Store-bandwidth-bound: 768 MB output -> ~33us floor @ 23.3 TB/s.
// Planes (1.5 MB) kept L2-resident; gathers are one aligned 128B line per
// wave per corner after a channel-last transpose. Output uses NT stores.

#define KP_RES   64
#define KP_FDIM  32
#define KP_HW    (KP_RES * KP_RES)          // 4096
#define KP_PLANE (KP_HW * KP_FDIM)          // 131072 floats = 512 KB
#define KP_TILE  256                        // points per block (8 waves)

// ---------------------------------------------------------------------------
// Kernel 1: transpose each plane [C=32, H=64, W=64] -> [H, W, C] into d_ws.
// Read once (NT loads); writes stay hot in L2 for the gather kernel.
// ---------------------------------------------------------------------------
__global__ void kp_transpose_kernel(const float* __restrict__ pxy,
                                    const float* __restrict__ pxz,
                                    const float* __restrict__ pyz,
                                    float* __restrict__ tp) {
  int idx = blockIdx.x * blockDim.x + threadIdx.x;
  if (idx >= 3 * KP_PLANE) return;
  int p  = idx >> 17;                // / 131072
  int r  = idx & (KP_PLANE - 1);
  int c  = r >> 12;                  // / 4096
  int hw = r & (KP_HW - 1);
  const float* __restrict__ src = (p == 0) ? pxy : (p == 1) ? pxz : pyz;
  float v = __builtin_nontemporal_load(&src[c * KP_HW + hw]);
  tp[p * KP_PLANE + hw * KP_FDIM + c] = v;   // scattered 128B-stride writes, tiny total
}

// ---------------------------------------------------------------------------
// Bilinear sample, channel-last plane, lane = channel (wave32 => 32 channels).
// Matches grid_sample(align_corners=True, padding_mode='border'): fx,fy in
// [0,63]; border handled by zero/+step offset selects whose weights are
// exactly 0 at the clamped edge.
// ---------------------------------------------------------------------------
__device__ __forceinline__ float kp_sample(const float* __restrict__ tpp,
                                           float fx, float fy, int lane) {
  int   x0 = (int)fx;
  int   y0 = (int)fy;
  float wx = fx - (float)x0;
  float wy = fy - (float)y0;
  int offx = (x0 < KP_RES - 1) ? KP_FDIM : 0;            // +1 in W = +32 floats
  int offy = (y0 < KP_RES - 1) ? KP_RES * KP_FDIM : 0;   // +1 in H = +2048 floats
  const float* bp = tpp + (y0 * (KP_RES * KP_FDIM) + x0 * KP_FDIM + lane);
  float v00 = bp[0];            // each load: one aligned 128B line per wave
  float v01 = bp[offx];
  float v10 = bp[offy];
  float v11 = bp[offx + offy];
  float top = v00 + wx * (v01 - v00);
  float bot = v10 + wx * (v11 - v10);
  return top + wy * (bot - top);
}

// ---------------------------------------------------------------------------
// Kernel 2: gather + interpolate. One block = 256-point tile. Coords are
// staged into LDS with CDNA5 async-to-LDS DMA, then each wave processes one
// point per iteration with lane = channel.
// ---------------------------------------------------------------------------
__global__ __launch_bounds__(KP_TILE)
void kp_gather_kernel(const float* __restrict__ xyz,
                      const float* __restrict__ tp,
                      float* __restrict__ out, int N) {
  __shared__ float sxyz[KP_TILE * 3];       // 3 KB coord tile
  const int tid      = threadIdx.x;
  const int tileBase = blockIdx.x * KP_TILE;

  if (tileBase + KP_TILE <= N) {
    // 768 floats = 192 lanes x 16B via async DMA straight into LDS.
    if (tid < (KP_TILE * 3) / 4) {
      unsigned lds = (unsigned)(size_t)(&sxyz[0]) + (unsigned)(tid * 16);
      unsigned long long ga =
          (unsigned long long)(const void*)(xyz + (size_t)tileBase * 3) +
          (unsigned long long)(tid * 16);
      asm volatile("global_load_async_to_lds_b128 %0, %1, off"
                   :: "v"(lds), "v"(ga) : "memory");
    }
    asm volatile("s_wait_asynccnt 0" ::: "memory");
  } else {
    // Tail tile: guarded plain staging (no OOB reads past xyz).
    for (int i = tid; i < KP_TILE * 3; i += KP_TILE) {
      long long g = (long long)tileBase * 3 + i;
      sxyz[i] = (g < (long long)N * 3) ? xyz[g] : 0.0f;
    }
  }
  __syncthreads();

  const int wave = tid >> 5;
  const int lane = tid & 31;

  for (int i = 0; i < 32; ++i) {
    int pt = (wave << 5) + i;
    int n  = tileBase + pt;
    if (n >= N) break;
    // Uniform-address LDS reads -> broadcast.
    float gx = sxyz[pt * 3 + 0];
    float gy = sxyz[pt * 3 + 1];
    float gz = sxyz[pt * 3 + 2];
    // BOUNDS=(-1,1): g = clip(xyz, -1, 1); coord = (g+1)*0.5*(RES-1).
    gx = fminf(fmaxf(gx, -1.f), 1.f);
    gy = fminf(fmaxf(gy, -1.f), 1.f);
    gz = fminf(fmaxf(gz, -1.f), 1.f);
    float fx = (gx + 1.f) * 31.5f;
    float fy = (gy + 1.f) * 31.5f;
    float fz = (gz + 1.f) * 31.5f;

    float r_xy = kp_sample(tp,                fx, fy, lane);  // (x->W, y->H)
    float r_xz = kp_sample(tp + KP_PLANE,     fx, fz, lane);
    float r_yz = kp_sample(tp + 2 * KP_PLANE, fy, fz, lane);

    // One contiguous 384B output row per point, streamed past L2 (NT).
    float* orow = out + (size_t)n * 96 + lane;
    __builtin_nontemporal_store(r_xy, orow);
    __builtin_nontemporal_store(r_xz, orow + 32);
    __builtin_nontemporal_store(r_yz, orow + 64);
  }
}

// ---------------------------------------------------------------------------
extern "C" void kernel_launch(void* const* d_in, const int* in_sizes, int n_in,
                              void* d_out, int out_size, void* d_ws, size_t ws_size,
                              hipStream_t stream) {
  const float* xyz = (const float*)d_in[0];
  const float* pxy = (const float*)d_in[1];
  const float* pxz = (const float*)d_in[2];
  const float* pyz = (const float*)d_in[3];
  float* out = (float*)d_out;
  float* tp  = (float*)d_ws;            // needs 3*131072*4 = 1.5 MB scratch

  int N = in_sizes[0] / 3;

  int tElems  = 3 * KP_PLANE;
  int tBlocks = (tElems + 255) / 256;
  kp_transpose_kernel<<<tBlocks, 256, 0, stream>>>(pxy, pxz, pyz, tp);

  int gBlocks = (N + KP_TILE - 1) / KP_TILE;
  kp_gather_kernel<<<gBlocks, KP_TILE, 0, stream>>>(xyz, tp, out, N);
}